// BeamSearchWithSrcInfer_55654186222148
// MI455X (gfx1250) — compile-verified
//
#include <hip/hip_runtime.h>
#include <math.h>

typedef float v2f __attribute__((ext_vector_type(2)));
typedef float v8f __attribute__((ext_vector_type(8)));

// D = A(16x4,f32) * B(4x16,f32) + C(16x16,f32), wave32 WMMA
#define WMMA_F32X4(a, b, c) \
  __builtin_amdgcn_wmma_f32_16x16x4_f32(false, (a), false, (b), (short)0, (c), false, false)

constexpr int S_  = 32;     // sequence steps
constexpr int B_  = 16;     // batch
constexpr int Hh  = 1024;   // hidden
constexpr int E_  = 256;    // embed
constexpr int V_  = 32000;  // vocab
constexpr int K_  = 4;      // beams
constexpr int DIN = E_ + E_ + Hh;   // 1536
constexpr int G4H = 4 * Hh;         // 4096
constexpr int R_  = K_ * B_;        // 64 rows (beam-major: r = k*B + b)
constexpr int START_TOK = 1;

// ---------------- init: replicate h0/c0 across beams, word=START ----------------
__global__ void init_state(const float* __restrict__ enc_h, const float* __restrict__ enc_c,
                           float* __restrict__ h, float* __restrict__ c, int* __restrict__ word) {
  int idx = blockIdx.x * blockDim.x + threadIdx.x;
  if (idx < R_ * Hh) {
    int r = idx / Hh, j = idx - r * Hh;
    int b = r & (B_ - 1);
    h[idx] = enc_h[b * Hh + j];
    c[idx] = enc_c[b * Hh + j];
  }
  if (idx < B_ * K_) word[idx] = START_TOK;
}

// ---------------- gather: X[r] = concat(embed[word], enc_in[t], enc_out[t]) ----------------
__global__ void gather_x(const int* __restrict__ word, const float* __restrict__ embed,
                         const float* __restrict__ enc_in, const float* __restrict__ enc_out,
                         float* __restrict__ X, int t) {
  int idx = blockIdx.x * blockDim.x + threadIdx.x;
  if (idx >= R_ * DIN) return;
  int r = idx / DIN, cc = idx - r * DIN;
  int b = r & (B_ - 1), k = r >> 4;
  float v;
  if (cc < E_) {
    int w = word[b * K_ + k];
    v = embed[(size_t)w * E_ + cc];
  } else if (cc < 2 * E_) {
    v = enc_in[(size_t)t * B_ * E_ + b * E_ + (cc - E_)];
  } else {
    v = enc_out[(size_t)t * B_ * Hh + b * Hh + (cc - 2 * E_)];
  }
  X[idx] = v;
}

// ---------------- gates GEMM: gates = X@w_ih + hprev@w_hh + b (WMMA f32) ----------------
// one wave per 16-wide N tile; 4 M-tiles accumulated per wave (all 64 rows)
__global__ void gates_gemm(const float* __restrict__ X, const float* __restrict__ hprev,
                           const float* __restrict__ w_ih, const float* __restrict__ w_hh,
                           const float* __restrict__ bl, float* __restrict__ gates) {
  int wid  = (blockIdx.x * blockDim.x + threadIdx.x) >> 5;
  int lane = threadIdx.x & 31;
  int half = lane >> 4, lcol = lane & 15;
  int n0 = wid * 16;
  float bias = bl[n0 + lcol];
  v8f acc[4];
#pragma unroll
  for (int m = 0; m < 4; ++m)
#pragma unroll
    for (int r = 0; r < 8; ++r) acc[m][r] = bias;

  for (int k = 0; k < DIN; k += 4) {
    v2f bf;
    bf.x = w_ih[(size_t)(k + 2 * half + 0) * G4H + n0 + lcol];
    bf.y = w_ih[(size_t)(k + 2 * half + 1) * G4H + n0 + lcol];
#pragma unroll
    for (int m = 0; m < 4; ++m) {
      int row = m * 16 + lcol;
      v2f af = *reinterpret_cast<const v2f*>(X + (size_t)row * DIN + k + 2 * half);
      acc[m] = WMMA_F32X4(af, bf, acc[m]);
    }
  }
  for (int k = 0; k < Hh; k += 4) {
    v2f bf;
    bf.x = w_hh[(size_t)(k + 2 * half + 0) * G4H + n0 + lcol];
    bf.y = w_hh[(size_t)(k + 2 * half + 1) * G4H + n0 + lcol];
#pragma unroll
    for (int m = 0; m < 4; ++m) {
      int row = m * 16 + lcol;
      v2f af = *reinterpret_cast<const v2f*>(hprev + (size_t)row * Hh + k + 2 * half);
      acc[m] = WMMA_F32X4(af, bf, acc[m]);
    }
  }
#pragma unroll
  for (int m = 0; m < 4; ++m)
#pragma unroll
    for (int r = 0; r < 8; ++r)
      gates[(size_t)(m * 16 + r + 8 * half) * G4H + n0 + lcol] = acc[m][r];
}

// ---------------- LSTM elementwise (in-place h,c) ----------------
__device__ __forceinline__ float sigm(float x) { return 1.f / (1.f + expf(-x)); }

__global__ void lstm_elem(const float* __restrict__ gates, float* __restrict__ h, float* __restrict__ c) {
  int idx = blockIdx.x * blockDim.x + threadIdx.x;
  if (idx >= R_ * Hh) return;
  int r = idx / Hh, j = idx - r * Hh;
  const float* g = gates + (size_t)r * G4H;
  float ig = sigm(g[j]);
  float fg = sigm(g[Hh + j]);
  float gg = tanhf(g[2 * Hh + j]);
  float og = sigm(g[3 * Hh + j]);
  float cn = fg * c[idx] + ig * gg;
  c[idx] = cn;
  h[idx] = og * tanhf(cn);
}

// ---------------- logits GEMM: logits = h@w_out + b_out (WMMA f32) ----------------
__global__ void logits_gemm(const float* __restrict__ h, const float* __restrict__ w_out,
                            const float* __restrict__ b_out, float* __restrict__ logits) {
  int wid  = (blockIdx.x * blockDim.x + threadIdx.x) >> 5;
  int lane = threadIdx.x & 31;
  int half = lane >> 4, lcol = lane & 15;
  int n0 = wid * 16;
  float bias = b_out[n0 + lcol];
  v8f acc[4];
#pragma unroll
  for (int m = 0; m < 4; ++m)
#pragma unroll
    for (int r = 0; r < 8; ++r) acc[m][r] = bias;

  for (int k = 0; k < Hh; k += 4) {
    v2f bf;
    bf.x = w_out[(size_t)(k + 2 * half + 0) * V_ + n0 + lcol];
    bf.y = w_out[(size_t)(k + 2 * half + 1) * V_ + n0 + lcol];
#pragma unroll
    for (int m = 0; m < 4; ++m) {
      int row = m * 16 + lcol;
      v2f af = *reinterpret_cast<const v2f*>(h + (size_t)row * Hh + k + 2 * half);
      acc[m] = WMMA_F32X4(af, bf, acc[m]);
    }
  }
#pragma unroll
  for (int m = 0; m < 4; ++m)
#pragma unroll
    for (int r = 0; r < 8; ++r)
      logits[(size_t)(m * 16 + r + 8 * half) * V_ + n0 + lcol] = acc[m][r];
}

// ---------------- per-row softmax stats ----------------
__global__ void row_stats(const float* __restrict__ logits, float* __restrict__ rowmax,
                          float* __restrict__ rowsum) {
  __shared__ float red[256];
  int r = blockIdx.x;
  const float* row = logits + (size_t)r * V_;
  float m = -INFINITY;
  for (int j = threadIdx.x; j < V_; j += blockDim.x) m = fmaxf(m, row[j]);
  red[threadIdx.x] = m;
  __syncthreads();
  for (int s = 128; s > 0; s >>= 1) {
    if (threadIdx.x < s) red[threadIdx.x] = fmaxf(red[threadIdx.x], red[threadIdx.x + s]);
    __syncthreads();
  }
  float rm = red[0];
  __syncthreads();
  float sum = 0.f;
  for (int j = threadIdx.x; j < V_; j += blockDim.x) sum += expf(row[j] - rm);
  red[threadIdx.x] = sum;
  __syncthreads();
  for (int s = 128; s > 0; s >>= 1) {
    if (threadIdx.x < s) red[threadIdx.x] += red[threadIdx.x + s];
    __syncthreads();
  }
  if (threadIdx.x == 0) { rowmax[r] = rm; rowsum[r] = red[0]; }
}

// ---------------- top-4 helpers (tie-break: lower index, like jax top_k) ----------------
__device__ __forceinline__ bool tk_better(float v1, int i1, float v2, int i2) {
  return (v1 > v2) || (v1 == v2 && i1 < i2);
}
__device__ __forceinline__ void tk_insert(float v, int i, float bv[4], int bi[4]) {
  if (!tk_better(v, i, bv[3], bi[3])) return;
  bv[3] = v; bi[3] = i;
#pragma unroll
  for (int p = 3; p > 0; --p) {
    if (tk_better(bv[p], bi[p], bv[p - 1], bi[p - 1])) {
      float tv = bv[p]; bv[p] = bv[p - 1]; bv[p - 1] = tv;
      int ti = bi[p]; bi[p] = bi[p - 1]; bi[p - 1] = ti;
    }
  }
}

// ---------------- step 0: top-4 of probs (beam-0 rows), set scores/word/out[:, :, 0] ------
__global__ void topk_step0(const float* __restrict__ logits, const float* __restrict__ rowmax,
                           const float* __restrict__ rowsum, float* __restrict__ scores,
                           int* __restrict__ word, int* __restrict__ out) {
  __shared__ float sv[1024];
  __shared__ int   si[1024];
  int b = blockIdx.x;
  int r = b;  // beam 0 row
  float rm = rowmax[r], inv = 1.f / rowsum[r];
  float bv[4] = {-INFINITY, -INFINITY, -INFINITY, -INFINITY};
  int   bi[4] = {0x7fffffff, 0x7fffffff, 0x7fffffff, 0x7fffffff};
  for (int j = threadIdx.x; j < V_; j += blockDim.x) {
    float v = expf(logits[(size_t)r * V_ + j] - rm) * inv;
    tk_insert(v, j, bv, bi);
  }
#pragma unroll
  for (int q = 0; q < 4; ++q) { sv[threadIdx.x * 4 + q] = bv[q]; si[threadIdx.x * 4 + q] = bi[q]; }
  __syncthreads();
  if (threadIdx.x == 0) {
    float fv[4] = {-INFINITY, -INFINITY, -INFINITY, -INFINITY};
    int   fi[4] = {0x7fffffff, 0x7fffffff, 0x7fffffff, 0x7fffffff};
    for (int p = 0; p < 1024; ++p) tk_insert(sv[p], si[p], fv, fi);
#pragma unroll
    for (int q = 0; q < 4; ++q) {
      scores[b * K_ + q] = fv[q];
      word[b * K_ + q]   = fi[q];
      out[q * B_ * S_ + b * S_ + 0] = fi[q];
    }
  }
}

// ---------------- steps 1..31: top-4 of probs*scores over K*V joint space ----------------
__global__ void topk_step(const float* __restrict__ logits, const float* __restrict__ rowmax,
                          const float* __restrict__ rowsum, const float* __restrict__ scores,
                          int* __restrict__ word, int* __restrict__ out, int t) {
  __shared__ float sv[1024];
  __shared__ int   si[1024];
  int b = blockIdx.x;
  float rm[K_], inv[K_], sc[K_];
#pragma unroll
  for (int k = 0; k < K_; ++k) {
    int r = k * B_ + b;
    rm[k] = rowmax[r]; inv[k] = 1.f / rowsum[r]; sc[k] = scores[b * K_ + k];
  }
  float bv[4] = {-INFINITY, -INFINITY, -INFINITY, -INFINITY};
  int   bi[4] = {0x7fffffff, 0x7fffffff, 0x7fffffff, 0x7fffffff};
  for (int i = threadIdx.x; i < K_ * V_; i += blockDim.x) {
    int k = i / V_, j = i - k * V_;
    float v = expf(logits[(size_t)(k * B_ + b) * V_ + j] - rm[k]) * inv[k] * sc[k];
    tk_insert(v, i, bv, bi);
  }
#pragma unroll
  for (int q = 0; q < 4; ++q) { sv[threadIdx.x * 4 + q] = bv[q]; si[threadIdx.x * 4 + q] = bi[q]; }
  __syncthreads();
  if (threadIdx.x == 0) {
    float fv[4] = {-INFINITY, -INFINITY, -INFINITY, -INFINITY};
    int   fi[4] = {0x7fffffff, 0x7fffffff, 0x7fffffff, 0x7fffffff};
    for (int p = 0; p < 1024; ++p) tk_insert(sv[p], si[p], fv, fi);
#pragma unroll
    for (int q = 0; q < 4; ++q) {
      int ti = fi[q];
      word[b * K_ + q] = ti % V_;               // word2 = ti % V
      out[q * B_ * S_ + b * S_ + t] = ti;       // raw joint-space index, like reference
    }
  }
}

// ---------------- host ----------------
extern "C" void kernel_launch(void* const* d_in, const int* in_sizes, int n_in,
                              void* d_out, int out_size, void* d_ws, size_t ws_size,
                              hipStream_t stream) {
  const float* enc_h_n     = (const float*)d_in[0];
  const float* enc_c_n     = (const float*)d_in[1];
  const float* enc_outputs = (const float*)d_in[2];
  const float* enc_inputs  = (const float*)d_in[3];
  const float* embed       = (const float*)d_in[4];
  const float* w_ih        = (const float*)d_in[5];
  const float* w_hh        = (const float*)d_in[6];
  const float* b_lstm      = (const float*)d_in[7];
  const float* w_out       = (const float*)d_in[8];
  const float* b_out       = (const float*)d_in[9];
  int* out = (int*)d_out;

  char* ws = (char*)d_ws;
  float* X      = (float*)(ws + 0);          // 64*1536*4  = 393216
  float* gates  = (float*)(ws + 393216);     // 64*4096*4  = 1048576
  float* h      = (float*)(ws + 1441792);    // 64*1024*4  = 262144
  float* c      = (float*)(ws + 1703936);    // 64*1024*4  = 262144
  float* logits = (float*)(ws + 1966080);    // 64*32000*4 = 8192000
  float* rowmax = (float*)(ws + 10158080);   // 256
  float* rowsum = (float*)(ws + 10158336);   // 256
  float* scores = (float*)(ws + 10158592);   // 256
  int*   word   = (int*)  (ws + 10158848);   // 256

  init_state<<<256, 256, 0, stream>>>(enc_h_n, enc_c_n, h, c, word);
  for (int t = 0; t < S_; ++t) {
    gather_x<<<(R_ * DIN) / 256, 256, 0, stream>>>(word, embed, enc_inputs, enc_outputs, X, t);
    gates_gemm<<<(G4H / 16) / 8, 256, 0, stream>>>(X, h, w_ih, w_hh, b_lstm, gates);
    lstm_elem<<<(R_ * Hh) / 256, 256, 0, stream>>>(gates, h, c);
    logits_gemm<<<(V_ / 16) / 8, 256, 0, stream>>>(h, w_out, b_out, logits);
    row_stats<<<R_, 256, 0, stream>>>(logits, rowmax, rowsum);
    if (t == 0)
      topk_step0<<<B_, 256, 0, stream>>>(logits, rowmax, rowsum, scores, word, out);
    else
      topk_step<<<B_, 256, 0, stream>>>(logits, rowmax, rowsum, scores, word, out, t);
  }
}